// MessagePassingLayer_14113262535303
// MI455X (gfx1250) — compile-verified
//
#include <hip/hip_runtime.h>

typedef __attribute__((ext_vector_type(2))) float v2f;
typedef __attribute__((ext_vector_type(8))) float v8f;

#define NUM_SRC 100000
#define NUM_DST 100000
#define IN_DIM 128
#define OUT_DIM 128

// ---------------------------------------------------------------------------
// GEMM1: M = relu(A(nrows,128) @ W(128,128) + bias), using V_WMMA_F32_16X16X4_F32
// Block = 256 threads (8 waves). Each wave: 16 rows x 128 cols output tile.
// W is pre-permuted into LDS in exact B-fragment order: [k0][nt][lane] -> v2f
//   lane L supplies B[kb][col], B[kb+1][col], kb = 4*k0 + 2*(L>>4), col = 16*nt + (L&15)
// ---------------------------------------------------------------------------
__global__ __launch_bounds__(256) void gemm_relu_128(
    const float* __restrict__ A, const float* __restrict__ W,
    const float* __restrict__ bias, float* __restrict__ Out, int nrows)
{
  __shared__ v2f bfrag[32 * 8 * 32];  // 64 KB

  const int t = threadIdx.x;
  for (int i = t; i < 32 * 8 * 32; i += 256) {
    int k0 = i >> 8, nt = (i >> 5) & 7, l = i & 31;
    int kb = k0 * 4 + ((l >> 4) << 1);
    int cc = nt * 16 + (l & 15);
    v2f v;
    v.x = W[kb * 128 + cc];
    v.y = W[(kb + 1) * 128 + cc];
    bfrag[i] = v;
  }
  __syncthreads();

  const int wave = t >> 5, lane = t & 31;
  const int rowBase = blockIdx.x * 128 + wave * 16;
  int arow = rowBase + (lane & 15);
  if (arow >= nrows) arow = nrows - 1;               // clamp loads; stores masked
  const float* aptr = A + (size_t)arow * 128 + ((lane >> 4) << 1);
  const int col = lane & 15;

  v8f c[8];
#pragma unroll
  for (int nt = 0; nt < 8; ++nt) {
    float b = bias[nt * 16 + col];
    v8f cv;
#pragma unroll
    for (int j = 0; j < 8; ++j) cv[j] = b;
    c[nt] = cv;
  }

  for (int k0 = 0; k0 < 32; ++k0) {
    v2f a = *(const v2f*)(aptr + k0 * 4);            // A[row][4k0 + 2*(L>>4) + {0,1}]
#pragma unroll
    for (int nt = 0; nt < 8; ++nt) {
      v2f b = bfrag[((k0 << 3) + nt) * 32 + lane];
      c[nt] = __builtin_amdgcn_wmma_f32_16x16x4_f32(
          false, a, false, b, (short)0, c[nt], false, false);
    }
  }

  // C/D layout: lane L, vgpr v -> row = rowBase + v + 8*(L>>4), col = 16*nt + (L&15)
  const int rhi = (lane >> 4) * 8;
#pragma unroll
  for (int v = 0; v < 8; ++v) {
    int r = rowBase + rhi + v;
    if (r < nrows) {
      float* orow = Out + (size_t)r * 128;
#pragma unroll
      for (int nt = 0; nt < 8; ++nt)
        orow[nt * 16 + col] = fmaxf(c[nt][v], 0.0f);
    }
  }
}

// ---------------------------------------------------------------------------
// GEMM2: Out = relu([Dst | Agg/clamp(Cnt,1)](nrows,256) @ W2(256,128) + b2)
// Two K-phases of 128 each, reusing the same 64 KB LDS fragment buffer.
// ---------------------------------------------------------------------------
__global__ __launch_bounds__(256) void gemm2_update(
    const float* __restrict__ Dst, const float* __restrict__ Agg,
    const float* __restrict__ Cnt, const float* __restrict__ W2,
    const float* __restrict__ b2, float* __restrict__ Out, int nrows)
{
  __shared__ v2f bfrag[32 * 8 * 32];

  const int t = threadIdx.x, wave = t >> 5, lane = t & 31;
  const int rowBase = blockIdx.x * 128 + wave * 16;
  int arow = rowBase + (lane & 15);
  if (arow >= nrows) arow = nrows - 1;
  const int col = lane & 15;
  const int klane = (lane >> 4) << 1;

  v8f c[8];
#pragma unroll
  for (int nt = 0; nt < 8; ++nt) {
    float b = b2[nt * 16 + col];
    v8f cv;
#pragma unroll
    for (int j = 0; j < 8; ++j) cv[j] = b;
    c[nt] = cv;
  }

  for (int phase = 0; phase < 2; ++phase) {
    const float* Wp = W2 + (size_t)phase * 128 * 128;
    __syncthreads();
    for (int i = t; i < 32 * 8 * 32; i += 256) {
      int k0 = i >> 8, nt = (i >> 5) & 7, l = i & 31;
      int kb = k0 * 4 + ((l >> 4) << 1);
      int cc = nt * 16 + (l & 15);
      v2f v;
      v.x = Wp[kb * 128 + cc];
      v.y = Wp[(kb + 1) * 128 + cc];
      bfrag[i] = v;
    }
    __syncthreads();

    const float* Ap;
    float scale = 1.0f;
    if (phase == 0) {
      Ap = Dst + (size_t)arow * 128 + klane;
    } else {
      Ap = Agg + (size_t)arow * 128 + klane;
      scale = 1.0f / fmaxf(Cnt[arow], 1.0f);          // fold scatter-mean divide in
    }

    for (int k0 = 0; k0 < 32; ++k0) {
      v2f a = *(const v2f*)(Ap + k0 * 4);
      a.x *= scale;
      a.y *= scale;
#pragma unroll
      for (int nt = 0; nt < 8; ++nt) {
        v2f b = bfrag[((k0 << 3) + nt) * 32 + lane];
        c[nt] = __builtin_amdgcn_wmma_f32_16x16x4_f32(
            false, a, false, b, (short)0, c[nt], false, false);
      }
    }
  }

  const int rhi = (lane >> 4) * 8;
#pragma unroll
  for (int v = 0; v < 8; ++v) {
    int r = rowBase + rhi + v;
    if (r < nrows) {
      float* orow = Out + (size_t)r * 128;
#pragma unroll
      for (int nt = 0; nt < 8; ++nt)
        orow[nt * 16 + col] = fmaxf(c[nt][v], 0.0f);
    }
  }
}

// ---------------------------------------------------------------------------
// Scatter: one wave per edge; lane i adds M[src][4i..4i+3] into agg[dst][...]
// ---------------------------------------------------------------------------
__global__ __launch_bounds__(256) void scatter_kernel(
    const float* __restrict__ M, const int* __restrict__ src_idx,
    const int* __restrict__ dst_idx, float* __restrict__ agg,
    float* __restrict__ cnt, int nedges)
{
  long long tid = (long long)blockIdx.x * blockDim.x + threadIdx.x;
  int e = (int)(tid >> 5);
  int c4 = (int)(tid & 31);
  if (e >= nedges) return;
  int s = src_idx[e];
  int d = dst_idx[e];
  const float4 mv = ((const float4*)(M + (size_t)s * 128))[c4];
  float* ar = agg + (size_t)d * 128 + c4 * 4;
  atomicAdd(ar + 0, mv.x);
  atomicAdd(ar + 1, mv.y);
  atomicAdd(ar + 2, mv.z);
  atomicAdd(ar + 3, mv.w);
  if (c4 == 0) atomicAdd(cnt + d, 1.0f);
}

// ---------------------------------------------------------------------------
extern "C" void kernel_launch(void* const* d_in, const int* in_sizes, int n_in,
                              void* d_out, int out_size, void* d_ws, size_t ws_size,
                              hipStream_t stream) {
  const float* src_emb = (const float*)d_in[0];
  const float* dst_emb = (const float*)d_in[1];
  const int*   edges   = (const int*)d_in[2];   // (2, E), int32
  const float* W1      = (const float*)d_in[3];
  const float* b1      = (const float*)d_in[4];
  const float* W2      = (const float*)d_in[5];
  const float* b2      = (const float*)d_in[6];

  const int nsrc   = in_sizes[0] / IN_DIM;
  const int ndst   = in_sizes[1] / IN_DIM;
  const int nedges = in_sizes[2] / 2;
  const int* src_idx = edges;
  const int* dst_idx = edges + nedges;

  float* M   = (float*)d_ws;                       // (nsrc, 128)
  float* agg = M + (size_t)nsrc * OUT_DIM;         // (ndst, 128)
  float* cnt = agg + (size_t)ndst * OUT_DIM;       // (ndst)

  // agg and cnt are contiguous: zero both every call (deterministic)
  hipMemsetAsync(agg, 0, ((size_t)ndst * OUT_DIM + ndst) * sizeof(float), stream);

  dim3 blk(256);
  // Per-source-node message MLP (algebraically equals per-edge MLP on gathers)
  gemm_relu_128<<<dim3((nsrc + 127) / 128), blk, 0, stream>>>(src_emb, W1, b1, M, nsrc);

  long long sthreads = (long long)nedges * 32;
  scatter_kernel<<<dim3((unsigned)((sthreads + 255) / 256)), blk, 0, stream>>>(
      M, src_idx, dst_idx, agg, cnt, nedges);

  gemm2_update<<<dim3((ndst + 127) / 128), blk, 0, stream>>>(
      dst_emb, agg, cnt, W2, b2, (float*)d_out, ndst);
}